// BiLSTMModel_90108413870166
// MI455X (gfx1250) — compile-verified
//
#include <hip/hip_runtime.h>
#include <math.h>

// ---------------- problem constants ----------------
#define Bn   64
#define Tn   128
#define BT   (Bn*Tn)          // 8192
#define En   1024
#define WDn  200
#define DIN  400              // 2*WD
#define Hn   200
#define G3   600              // 3*H
#define G4   800              // 4*H
#define TAGn 5
#define BIG_NEG 1000000.0f

typedef __attribute__((ext_vector_type(2))) float v2f;
typedef __attribute__((ext_vector_type(8))) float v8f;

__device__ __forceinline__ float sigm(float x) { return 1.0f / (1.0f + __expf(-x)); }

// ================= K0: x_rel gather  =================
__global__ void k_rel_gather(const int* __restrict__ rels,
                             const float* __restrict__ rel_table,
                             float* __restrict__ xlex) {
  int idx = blockIdx.x * blockDim.x + threadIdx.x;  // BT*WD
  if (idx >= BT * WDn) return;
  int m = idx / WDn, o = idx % WDn;
  xlex[(long long)m * DIN + WDn + o] = rel_table[rels[m] * WDn + o];
}

// ================= K1: fused elmo-mix + embed GEMM (WMMA f32 16x16x4) =========
// C[m][n] = sum_k (sum_l w_l * elmos[b,l,t,k]) * W_elmo[n][k],  m=b*T+t
// Each wave computes one 16-row M tile x two 16-col N tiles (A fragment reused).
// grid = (ceil(200/32)=7, 8192/16=512), block = 32.
__global__ void k_elmo_gemm(const float* __restrict__ elmos,
                            const float* __restrict__ ew,
                            const float* __restrict__ W,
                            float* __restrict__ xlex) {
  int lane = threadIdx.x & 31;
  int r = lane & 15, hi = lane >> 4;
  int nt = blockIdx.x, mt = blockIdx.y;
  int m = mt * 16 + r;          // A-fragment row
  int b = m >> 7;               // m / T
  int t = m & 127;              // m % T
  const float* __restrict__ pa = elmos + (long long)b * 3LL * Tn * En + (long long)t * En;
  int n0 = nt * 32 + r;
  int n1 = n0 + 16;
  // clamp OOB weight rows to row 0; their accumulators are never stored
  const float* __restrict__ pw0 = W + (long long)(n0 < WDn ? n0 : 0) * En;
  const float* __restrict__ pw1 = W + (long long)(n1 < WDn ? n1 : 0) * En;
  const float w0 = ew[0], w1 = ew[1], w2 = ew[2];
  __builtin_prefetch(pa, 0, 0);
  __builtin_prefetch(pw0, 0, 0);
  v8f c0 = {}, c1 = {};
  for (int kk = 0; kk < En; kk += 4) {
    int k0 = kk + 2 * hi;
    v2f e0 = *(const v2f*)(pa + k0);
    v2f e1 = *(const v2f*)(pa + Tn * En + k0);
    v2f e2 = *(const v2f*)(pa + 2 * Tn * En + k0);
    v2f a;
    a.x = w0 * e0.x + w1 * e1.x + w2 * e2.x;
    a.y = w0 * e0.y + w1 * e1.y + w2 * e2.y;
    v2f bb0 = *(const v2f*)(pw0 + k0);
    v2f bb1 = *(const v2f*)(pw1 + k0);
    c0 = __builtin_amdgcn_wmma_f32_16x16x4_f32(false, a, false, bb0, (short)0, c0, false, false);
    c1 = __builtin_amdgcn_wmma_f32_16x16x4_f32(false, a, false, bb1, (short)0, c1, false, false);
  }
  if (n0 < WDn) {
    #pragma unroll
    for (int v = 0; v < 8; ++v)
      xlex[(long long)(mt * 16 + v + 8 * hi) * DIN + n0] = c0[v];
  }
  if (n1 < WDn) {
    #pragma unroll
    for (int v = 0; v < 8; ++v)
      xlex[(long long)(mt * 16 + v + 8 * hi) * DIN + n1] = c1[v];
  }
}

// ================= K2: generic C = A @ W^T + bias  (WMMA f32 16x16x4) =========
// A: M x K (lda), W: N x K, C: M x ldc.  Two N tiles per wave (A reused).
// grid = (ceil(N/32), M/16), block = 32.
__global__ void k_gemm_bias(const float* __restrict__ A, int lda,
                            const float* __restrict__ W, int K,
                            const float* __restrict__ bias,
                            float* __restrict__ C, int ldc, int N) {
  int lane = threadIdx.x & 31;
  int r = lane & 15, hi = lane >> 4;
  int nt = blockIdx.x, mt = blockIdx.y;
  int arow = mt * 16 + r;
  int n0 = nt * 32 + r;
  int n1 = n0 + 16;
  const float* __restrict__ Ap  = A + (long long)arow * lda;
  const float* __restrict__ Wp0 = W + (long long)(n0 < N ? n0 : 0) * K;
  const float* __restrict__ Wp1 = W + (long long)(n1 < N ? n1 : 0) * K;
  float b0 = bias[n0 < N ? n0 : 0];
  float b1 = bias[n1 < N ? n1 : 0];
  __builtin_prefetch(Ap, 0, 0);
  __builtin_prefetch(Wp0, 0, 0);
  v8f c0 = {b0, b0, b0, b0, b0, b0, b0, b0};
  v8f c1 = {b1, b1, b1, b1, b1, b1, b1, b1};
  for (int kk = 0; kk < K; kk += 4) {
    int k0 = kk + 2 * hi;
    v2f a   = *(const v2f*)(Ap + k0);
    v2f bb0 = *(const v2f*)(Wp0 + k0);
    v2f bb1 = *(const v2f*)(Wp1 + k0);
    c0 = __builtin_amdgcn_wmma_f32_16x16x4_f32(false, a, false, bb0, (short)0, c0, false, false);
    c1 = __builtin_amdgcn_wmma_f32_16x16x4_f32(false, a, false, bb1, (short)0, c1, false, false);
  }
  if (n0 < N) {
    #pragma unroll
    for (int v = 0; v < 8; ++v)
      C[(long long)(mt * 16 + v + 8 * hi) * ldc + n0] = c0[v];
  }
  if (n1 < N) {
    #pragma unroll
    for (int v = 0; v < 8; ++v)
      C[(long long)(mt * 16 + v + 8 * hi) * ldc + n1] = c1[v];
  }
}

// ================= K3: zero scratch =================
__global__ void k_zero(float* __restrict__ p, int n) {
  int i = blockIdx.x * blockDim.x + threadIdx.x;
  if (i < n) p[i] = 0.f;
}

// ================= K4: bottom-up tree GRU scan (one block per batch) =========
__global__ void k_dt_scan(const float* __restrict__ gi, const float* __restrict__ Whh,
                          const float* __restrict__ bhh, const int* __restrict__ heads,
                          float* __restrict__ acc, float* __restrict__ tree_out) {
  int b = blockIdx.x, tid = threadIdx.x;   // 256 threads
  __shared__ float gh_s[G3];
  for (int j = Tn - 1; j >= 0; --j) {
    const float* accj = acc + (long long)(b * Tn + j) * Hn;
    for (int o = tid; o < G3; o += 256) {
      float s = bhh[o];
      const float* wr = Whh + (long long)o * Hn;
      #pragma unroll 4
      for (int k = 0; k < Hn; ++k) s += accj[k] * wr[k];
      gh_s[o] = s;
    }
    __syncthreads();
    int p = heads[b * Tn + j];
    if (tid < Hn) {
      const float* g = gi + (long long)(b * Tn + j) * G3;
      float r = sigm(g[tid] + gh_s[tid]);
      float z = sigm(g[Hn + tid] + gh_s[Hn + tid]);
      float n = tanhf(g[2 * Hn + tid] + r * gh_s[2 * Hn + tid]);
      float hp = accj[tid];
      float h = (1.f - z) * n + z * hp;
      tree_out[(long long)(b * Tn + j) * DIN + tid] = h;           // cols [0,200)
      if (p >= 0) acc[(long long)(b * Tn + p) * Hn + tid] += h;    // scatter to parent
    }
    __syncthreads();
  }
}

// ================= K5: top-down tree GRU scan =================
__global__ void k_td_scan(const float* __restrict__ gi, const float* __restrict__ Whh,
                          const float* __restrict__ bhh, const int* __restrict__ heads,
                          float* __restrict__ tree_out) {
  int b = blockIdx.x, tid = threadIdx.x;
  __shared__ float hp_s[Hn];
  __shared__ float gh_s[G3];
  for (int j = 0; j < Tn; ++j) {
    int p = heads[b * Tn + j];
    if (tid < Hn)
      hp_s[tid] = (p >= 0) ? tree_out[(long long)(b * Tn + p) * DIN + Hn + tid] : 0.f;
    __syncthreads();
    for (int o = tid; o < G3; o += 256) {
      float s = bhh[o];
      const float* wr = Whh + (long long)o * Hn;
      #pragma unroll 4
      for (int k = 0; k < Hn; ++k) s += hp_s[k] * wr[k];
      gh_s[o] = s;
    }
    __syncthreads();
    if (tid < Hn) {
      const float* g = gi + (long long)(b * Tn + j) * G3;
      float r = sigm(g[tid] + gh_s[tid]);
      float z = sigm(g[Hn + tid] + gh_s[Hn + tid]);
      float n = tanhf(g[2 * Hn + tid] + r * gh_s[2 * Hn + tid]);
      float h = (1.f - z) * n + z * hp_s[tid];
      tree_out[(long long)(b * Tn + j) * DIN + Hn + tid] = h;      // cols [200,400)
    }
    __syncthreads();
  }
}

// ================= K6: LSTM scan, 16 batches/block, recurrent matmul via WMMA =
// dynamic LDS: h[16*200] | c[16*200] | gh[16*800]  = 76800 bytes
template <int FWD>
__global__ void k_lstm_scan(const float* __restrict__ gi, const float* __restrict__ Whh,
                            const float* __restrict__ masks, float* __restrict__ hout) {
  extern __shared__ float smem[];
  float* h_s  = smem;                 // 16*200
  float* c_s  = smem + 16 * Hn;       // 16*200
  float* gh_s = smem + 32 * Hn;       // 16*800
  int tid = threadIdx.x;              // 256 = 8 waves
  int lane = tid & 31, wave = tid >> 5;
  int r = lane & 15, hi = lane >> 4;
  int bstart = blockIdx.x * 16;
  for (int i = tid; i < 16 * Hn; i += 256) { h_s[i] = 0.f; c_s[i] = 0.f; }
  __syncthreads();
  for (int step = 0; step < Tn; ++step) {
    int t = FWD ? step : (Tn - 1 - step);
    // gh(16x800) = h(16x200) @ Whh^T(200x800), 50 N-tiles, K=200
    for (int nt = wave; nt < G4 / 16; nt += 8) {
      int wrow = nt * 16 + r;
      const float* wp = Whh + (long long)wrow * Hn;
      const float* hp = h_s + r * Hn;          // A row = local batch r
      v8f c = {};
      for (int kk = 0; kk < Hn; kk += 4) {
        int k0 = kk + 2 * hi;
        v2f a, bb;
        a.x = hp[k0];  a.y = hp[k0 + 1];
        bb = *(const v2f*)(wp + k0);
        c = __builtin_amdgcn_wmma_f32_16x16x4_f32(false, a, false, bb, (short)0, c, false, false);
      }
      #pragma unroll
      for (int v = 0; v < 8; ++v) gh_s[(v + 8 * hi) * G4 + wrow] = c[v];
    }
    __syncthreads();
    for (int idx = tid; idx < 16 * Hn; idx += 256) {
      int lb = idx / Hn, o = idx % Hn;
      int b = bstart + lb;
      const float* g  = gi + (long long)(b * Tn + t) * G4;
      const float* gh = gh_s + (long long)lb * G4;
      float i_ = sigm(g[o]          + gh[o]);
      float f_ = sigm(g[Hn + o]     + gh[Hn + o]);
      float gg = tanhf(g[2 * Hn + o] + gh[2 * Hn + o]);
      float o_ = sigm(g[3 * Hn + o] + gh[3 * Hn + o]);
      float cprev = c_s[idx], hprev = h_s[idx];
      float cn = f_ * cprev + i_ * gg;
      float hn = o_ * tanhf(cn);
      float mm = masks[b * Tn + t];
      float hcur = mm * hn + (1.f - mm) * hprev;
      float ccur = mm * cn + (1.f - mm) * cprev;
      h_s[idx] = hcur; c_s[idx] = ccur;
      hout[(long long)(b * Tn + t) * Hn + o] = hcur;
    }
    __syncthreads();
  }
}

// ================= K7: masked max-pool + projection =================
__global__ void k_pool_proj(const float* __restrict__ hf, const float* __restrict__ hb,
                            const float* __restrict__ masks, const float* __restrict__ Wproj,
                            float* __restrict__ out) {
  int b = blockIdx.x, tid = threadIdx.x;   // 256 threads
  __shared__ float pooled[2 * Hn];
  for (int o = tid; o < 2 * Hn; o += 256) {
    const float* src = (o < Hn) ? (hf + (long long)b * Tn * Hn + o)
                                : (hb + (long long)b * Tn * Hn + (o - Hn));
    float mx = -INFINITY;
    for (int t = 0; t < Tn; ++t) {
      float v = src[(long long)t * Hn] + (masks[b * Tn + t] - 1.f) * BIG_NEG;
      mx = fmaxf(mx, v);
    }
    pooled[o] = mx;
  }
  __syncthreads();
  if (tid < TAGn) {
    float s = 0.f;
    const float* w = Wproj + tid * 2 * Hn;
    for (int k = 0; k < 2 * Hn; ++k) s += pooled[k] * w[k];
    out[b * TAGn + tid] = s;
  }
}

// ================= host-side orchestration =================
extern "C" void kernel_launch(void* const* d_in, const int* in_sizes, int n_in,
                              void* d_out, int out_size, void* d_ws, size_t ws_size,
                              hipStream_t stream) {
  const float* elmos    = (const float*)d_in[0];
  const int*   rels     = (const int*)  d_in[1];
  const int*   heads    = (const int*)  d_in[2];
  const float* masks    = (const float*)d_in[3];
  // d_in[4] = lengths (unused)
  const float* ew       = (const float*)d_in[5];
  const float* W_elmo   = (const float*)d_in[6];
  const float* rel_tab  = (const float*)d_in[7];
  const float* dt_Wih   = (const float*)d_in[8];
  const float* dt_Whh   = (const float*)d_in[9];
  const float* dt_bih   = (const float*)d_in[10];
  const float* dt_bhh   = (const float*)d_in[11];
  const float* td_Wih   = (const float*)d_in[12];
  const float* td_Whh   = (const float*)d_in[13];
  const float* td_bih   = (const float*)d_in[14];
  const float* td_bhh   = (const float*)d_in[15];
  const float* lf_Wih   = (const float*)d_in[16];
  const float* lf_Whh   = (const float*)d_in[17];
  const float* lf_b     = (const float*)d_in[18];
  const float* lb_Wih   = (const float*)d_in[19];
  const float* lb_Whh   = (const float*)d_in[20];
  const float* lb_b     = (const float*)d_in[21];
  const float* W_proj   = (const float*)d_in[22];
  float* out = (float*)d_out;

  // workspace layout (floats), lifetime-aliased
  float* ws = (float*)d_ws;
  const long long SZ_XLEX = (long long)BT * DIN;   // 3,276,800
  const long long SZ_GI3  = (long long)BT * G3;    // 4,915,200
  const long long SZ_GI4  = (long long)BT * G4;    // 6,553,600
  const long long SZ_TREE = (long long)BT * DIN;
  const long long SZ_ACC  = (long long)BT * Hn;
  const long long SZ_H    = (long long)BT * Hn;
  float* x_lex = ws;                                   // region A start
  float* gi_dt = ws + SZ_XLEX;
  float* gi_td = gi_dt + SZ_GI3;
  float* gi_lf = ws;                                   // reuses x_lex/gi_dt
  float* gi_lb = ws + SZ_GI4;                          // reuses gi_td tail
  float* tree  = ws + SZ_XLEX + 2 * SZ_GI3;            // after region A
  float* accb  = tree + SZ_TREE;
  float* hf    = accb + SZ_ACC;
  float* hb    = hf + SZ_H;

  // 1) x_rel gather into x_lex[:,200:400]
  k_rel_gather<<<dim3((BT * WDn + 255) / 256), dim3(256), 0, stream>>>(rels, rel_tab, x_lex);

  // 2) fused elmo mix + embed GEMM into x_lex[:,0:200]
  k_elmo_gemm<<<dim3((WDn + 31) / 32, BT / 16), dim3(32), 0, stream>>>(elmos, ew, W_elmo, x_lex);

  // 3) hoisted GRU input-gate GEMMs
  k_gemm_bias<<<dim3((G3 + 31) / 32, BT / 16), dim3(32), 0, stream>>>(
      x_lex, DIN, dt_Wih, DIN, dt_bih, gi_dt, G3, G3);
  k_gemm_bias<<<dim3((G3 + 31) / 32, BT / 16), dim3(32), 0, stream>>>(
      x_lex, DIN, td_Wih, DIN, td_bih, gi_td, G3, G3);

  // 4) tree scans (acc zero-init, then bottom-up, then top-down)
  k_zero<<<dim3((int)((SZ_ACC + 255) / 256)), dim3(256), 0, stream>>>(accb, (int)SZ_ACC);
  k_dt_scan<<<dim3(Bn), dim3(256), 0, stream>>>(gi_dt, dt_Whh, dt_bhh, heads, accb, tree);
  k_td_scan<<<dim3(Bn), dim3(256), 0, stream>>>(gi_td, td_Whh, td_bhh, heads, tree);

  // 5) hoisted LSTM input-gate GEMMs (region A reuse is safe: gi_dt/gi_td dead)
  k_gemm_bias<<<dim3((G4 + 31) / 32, BT / 16), dim3(32), 0, stream>>>(
      tree, DIN, lf_Wih, DIN, lf_b, gi_lf, G4, G4);
  k_gemm_bias<<<dim3((G4 + 31) / 32, BT / 16), dim3(32), 0, stream>>>(
      tree, DIN, lb_Wih, DIN, lb_b, gi_lb, G4, G4);

  // 6) BiLSTM scans (WMMA recurrent matmul), dynamic LDS = 76800 B
  const size_t lstm_lds = (size_t)(16 * Hn * 2 + 16 * G4) * sizeof(float);
  k_lstm_scan<1><<<dim3(Bn / 16), dim3(256), lstm_lds, stream>>>(gi_lf, lf_Whh, masks, hf);
  k_lstm_scan<0><<<dim3(Bn / 16), dim3(256), lstm_lds, stream>>>(gi_lb, lb_Whh, masks, hb);

  // 7) masked max-pool + projection
  k_pool_proj<<<dim3(Bn), dim3(256), 0, stream>>>(hf, hb, masks, W_proj, out);
}